// FocalLoss_41360535060961
// MI455X (gfx1250) — compile-verified
//
#include <hip/hip_runtime.h>
#include <math.h>

#define BATCH 2
#define NREF 3
#define FDIM 256
#define HW 4096
#define NROWS (NREF * HW)          // 12288 reference pixels
#define ROW_TILES (NROWS / 16)     // 768
#define NCLS 10
#define COLS_PER_WG 64
#define NWAVES 8

typedef __attribute__((ext_vector_type(2))) float v2f;
typedef __attribute__((ext_vector_type(8))) float v8f;

__global__ void fl_zero_ws(float* w) {
    if (threadIdx.x < 2) w[threadIdx.x] = 0.0f;
}

__global__ __launch_bounds__(256) void fl_main(
    const float* __restrict__ ref,        // (B, NREF, FDIM, H, W)
    const float* __restrict__ target,     // (B, FDIM, H, W)
    const float* __restrict__ ref_label,  // (B, NREF, NCLS, H, W) one-hot
    const int*   __restrict__ tgt_label,  // (B, H, W)
    float* __restrict__ wsum)             // [0]=sum log_pt, [1]=sum focal
{
    // Bs[cc][c16][k] padded to 260 so ds_load_b64 B-fragments are bank-conflict free
    __shared__ float Bs[4][16][260];      // 66,560 B (reused as merge buffers later)
    __shared__ float Etile[NWAVES][16][20];

    const int tid    = threadIdx.x;
    const int lane   = tid & 31;
    const int wv     = tid >> 5;
    const int lane16 = lane & 15;
    const int half   = lane >> 4;
    const int koff   = half * 2;          // K offset for A/B fragments (WMMA layout)

    const int b     = blockIdx.x / (HW / COLS_PER_WG);
    const int tbase = (blockIdx.x % (HW / COLS_PER_WG)) * COLS_PER_WG;

    // --- stage target tile (FDIM x 64 cols) into LDS ---
    for (int idx = tid; idx < FDIM * COLS_PER_WG; idx += 256) {
        const int k = idx >> 6;
        const int c = idx & 63;
        Bs[c >> 4][c & 15][k] = target[(b * FDIM + k) * HW + tbase + c];
    }
    __syncthreads();

    const v8f zero8 = {0.f, 0.f, 0.f, 0.f, 0.f, 0.f, 0.f, 0.f};
    v8f   Pacc[4];                        // per-class accumulators (C-layout)
    float mold[4];
    #pragma unroll
    for (int cc = 0; cc < 4; ++cc) { Pacc[cc] = zero8; mold[cc] = -INFINITY; }

    for (int rt = wv; rt < ROW_TILES; rt += NWAVES) {
        const int r0  = rt * 16;
        const int n   = r0 >> 12;         // reference index (tile never crosses)
        const int hw0 = r0 & 4095;
        const float* aBase = ref + ((b * NREF + n) * FDIM) * HW + hw0 + lane16;

        // ---------- GEMM1: sim tile (16 rows x 64 cols), K = 256 ----------
        v8f acc[4];
        #pragma unroll
        for (int cc = 0; cc < 4; ++cc) acc[cc] = zero8;

        for (int kc = 0; kc < FDIM / 4; ++kc) {
            const int k = kc * 4 + koff;
            v2f A;
            A.x = aBase[k * HW];          // coalesced: 16 lanes -> 16 contiguous hw
            A.y = aBase[(k + 1) * HW];
            #pragma unroll
            for (int cc = 0; cc < 4; ++cc) {
                const v2f Bf = *(const v2f*)&Bs[cc][lane16][k];   // ds_load_b64
                acc[cc] = __builtin_amdgcn_wmma_f32_16x16x4_f32(
                    false, A, false, Bf, (short)0, acc[cc], false, false);
            }
        }

        // ---------- labels of this tile's 16 rows (both halves redundant) ----------
        int mylbl = 0;
        {
            const float* lb = ref_label + ((b * NREF + n) * NCLS) * HW + hw0 + lane16;
            #pragma unroll
            for (int d = 0; d < NCLS; ++d)
                mylbl = (lb[d * HW] > 0.5f) ? d : mylbl;
        }
        // one-hot A fragments for the class GEMM (A: class(M=16) x rows(K=4))
        v2f Af[4];
        #pragma unroll
        for (int kk = 0; kk < 4; ++kk) {
            const int la = __shfl(mylbl, 4 * kk + koff, 32);
            const int lb2 = __shfl(mylbl, 4 * kk + koff + 1, 32);
            Af[kk].x = (la  == lane16) ? 1.0f : 0.0f;
            Af[kk].y = (lb2 == lane16) ? 1.0f : 0.0f;
        }

        // ---------- online softmax + one-hot class GEMM ----------
        const int erow = half * 8;
        #pragma unroll
        for (int cc = 0; cc < 4; ++cc) {
            float lm = acc[cc][0];
            #pragma unroll
            for (int j = 1; j < 8; ++j) lm = fmaxf(lm, acc[cc][j]);
            lm = fmaxf(lm, __shfl_xor(lm, 16, 32));       // both halves share column
            const float mnew = fmaxf(mold[cc], lm);
            const float sc   = __expf(mold[cc] - mnew);   // 0 on first tile
            mold[cc] = mnew;
            #pragma unroll
            for (int j = 0; j < 8; ++j) Pacc[cc][j] *= sc;
            #pragma unroll
            for (int j = 0; j < 8; ++j)                   // C-layout -> LDS
                Etile[wv][erow + j][lane16] = __expf(acc[cc][j] - mnew);
            #pragma unroll
            for (int kk = 0; kk < 4; ++kk) {              // reload in B-layout
                v2f Ef;
                Ef.x = Etile[wv][4 * kk + koff][lane16];
                Ef.y = Etile[wv][4 * kk + koff + 1][lane16];
                Pacc[cc] = __builtin_amdgcn_wmma_f32_16x16x4_f32(
                    false, Af[kk], false, Ef, (short)0, Pacc[cc], false, false);
            }
        }
    }

    // ---------- cross-wave merge (overlay buffers on Bs, now dead) ----------
    __syncthreads();
    float* mBuf = &Bs[0][0][0];                             // [NWAVES][64]
    float* PBuf = &Bs[0][0][0] + NWAVES * COLS_PER_WG;      // [NWAVES][64][12]

    #pragma unroll
    for (int cc = 0; cc < 4; ++cc) {
        const int col = cc * 16 + lane16;
        if (half == 0) mBuf[wv * COLS_PER_WG + col] = mold[cc];
        #pragma unroll
        for (int j = 0; j < 8; ++j) {
            const int cls = j + half * 8;
            if (cls < NCLS)
                PBuf[(wv * COLS_PER_WG + col) * 12 + cls] = Pacc[cc][j];
        }
    }
    __syncthreads();

    if (tid < COLS_PER_WG) {
        const int col = tid;
        float M = -INFINITY;
        for (int w = 0; w < NWAVES; ++w)
            M = fmaxf(M, mBuf[w * COLS_PER_WG + col]);
        float P[NCLS];
        #pragma unroll
        for (int d = 0; d < NCLS; ++d) P[d] = 0.0f;
        for (int w = 0; w < NWAVES; ++w) {
            const float s = __expf(mBuf[w * COLS_PER_WG + col] - M);
            #pragma unroll
            for (int d = 0; d < NCLS; ++d)
                P[d] += PBuf[(w * COLS_PER_WG + col) * 12 + d] * s;
        }
        float S = 0.0f;
        #pragma unroll
        for (int d = 0; d < NCLS; ++d) S += P[d];
        const float invS = 1.0f / S;                 // pred_d = P_d / S (sums to 1)
        float zmax = -INFINITY;
        #pragma unroll
        for (int d = 0; d < NCLS; ++d) { P[d] *= invS; zmax = fmaxf(zmax, P[d]); }
        float se = 0.0f;
        #pragma unroll
        for (int d = 0; d < NCLS; ++d) se += __expf(P[d] - zmax);
        const float lse = zmax + __logf(se);         // log-softmax over pred
        const int   tl  = tgt_label[b * HW + tbase + col];
        const float logpt = P[tl] - lse;
        const float focal = sqrtf(fmaxf(0.0f, 1.0f - __expf(logpt)));  // gamma=0.5
        atomicAdd(&wsum[0], logpt);
        atomicAdd(&wsum[1], focal);
    }
}

__global__ void fl_finalize(const float* __restrict__ wsum, float* __restrict__ out) {
    if (threadIdx.x == 0 && blockIdx.x == 0) {
        const float N  = (float)(BATCH * HW);
        const float ce = -wsum[0] / N;               // scalar NLL mean
        out[0] = ce * (wsum[1] / N);                 // mean(focal * ce)
    }
}

extern "C" void kernel_launch(void* const* d_in, const int* in_sizes, int n_in,
                              void* d_out, int out_size, void* d_ws, size_t ws_size,
                              hipStream_t stream) {
    const float* ref       = (const float*)d_in[0];
    const float* target    = (const float*)d_in[1];
    const float* ref_label = (const float*)d_in[2];
    const int*   tgt_label = (const int*)d_in[3];
    float* wsum = (float*)d_ws;
    float* out  = (float*)d_out;

    fl_zero_ws<<<1, 32, 0, stream>>>(wsum);
    fl_main<<<BATCH * (HW / COLS_PER_WG), 256, 0, stream>>>(
        ref, target, ref_label, tgt_label, wsum);
    fl_finalize<<<1, 32, 0, stream>>>(wsum, out);
}